// multi_dRNN_with_dilations_62646392980179
// MI455X (gfx1250) — compile-verified
//
#include <hip/hip_runtime.h>

// Problem constants (from reference): T=512, B=128, H=256, D_IN=256, L=3, dil={1,2,4}
#define T_LEN  512
#define B_SZ   128
#define H_SZ   256
#define G4H    1024           // 4*H
#define NLAYER 3
#define MAT_ELEMS (256 * 1024)  // one packed weight matrix (K x 4H)

typedef __attribute__((ext_vector_type(2))) float v2f;
typedef __attribute__((ext_vector_type(8))) float v8f;

#define LOG2E  1.44269504088896340f

// sigmoid(x) = rcp(1 + 2^(-x*log2 e))      -- v_exp_f32 + v_rcp_f32, no divide
__device__ __forceinline__ float fast_sigmoid(float x) {
  return __builtin_amdgcn_rcpf(1.0f + __builtin_amdgcn_exp2f(-LOG2E * x));
}
// tanh(x) = 1 - 2*rcp(2^(2x*log2 e) + 1)   -- v_exp_f32 + v_rcp_f32 + fma
__device__ __forceinline__ float fast_tanh(float x) {
  float r = __builtin_amdgcn_rcpf(__builtin_amdgcn_exp2f(2.0f * LOG2E * x) + 1.0f);
  return __builtin_fmaf(-2.0f, r, 1.0f);
}

// ---------------------------------------------------------------------------
// Pack Wih (4H x K) and Whh (4H x K) into WMMA-B-friendly layout:
//   WP[p][row][half] = W[row][2p + half]   (p = k/2 pair index)
// so a single 8-byte load per lane yields B[k..k+1][n] for v_wmma_f32_16x16x4.
// Six matrices (3 layers x {Wih,Whh}) -> 6 MB, L2-resident for the whole run.
// ---------------------------------------------------------------------------
__global__ void wt_pack_kernel(const float* __restrict__ Wih,
                               const float* __restrict__ Whh,
                               float* __restrict__ wp) {
  int gid  = blockIdx.x * blockDim.x + threadIdx.x;   // 0 .. 6*262144-1
  int mtx  = gid >> 18;                                // which matrix
  int o    = gid & (MAT_ELEMS - 1);
  int half = o & 1;
  int row  = (o >> 1) & 1023;
  int p    = o >> 11;
  int k    = 2 * p + half;
  int l    = mtx >> 1;
  const float* src = ((mtx & 1) == 0) ? (Wih + (size_t)l * G4H * 256)
                                      : (Whh + (size_t)l * G4H * 256);
  wp[gid] = src[row * 256 + k];
}

// ---------------------------------------------------------------------------
// Fused dilated-LSTM scan for one layer.
// Block: 512 threads = 16 waves; one block owns 16 packed-batch rows and runs
// the whole Td-step recurrence internally (no grid sync). Wave w owns
// h-columns [16w,16w+16) and all 4 gates for them, so the cell state c stays
// in that wave's VGPRs. h lives in LDS; x tiles are DMA'd into LDS with
// global_load_async_to_lds_b64 (ASYNCcnt). GEMMs use v_wmma_f32_16x16x4_f32.
// ---------------------------------------------------------------------------
__global__ __launch_bounds__(512, 1)
void dlstm_scan_kernel(const float* __restrict__ xin,   // (T,B,256) layer input
                       const float* __restrict__ wpx,   // packed Wih
                       const float* __restrict__ wph,   // packed Whh
                       const float* __restrict__ bih,
                       const float* __restrict__ bhh,
                       float* __restrict__ out,         // (T,B,H) layer output
                       int rate) {
  __shared__ float x_lds[16][258];   // +2 pad: rotates LDS banks per row
  __shared__ float h_lds[16][258];

  const int tid  = threadIdx.x;
  const int lane = tid & 31;
  const int wave = tid >> 5;
  const int lo   = lane & 15;     // A row / B,C column within the 16x16 tile
  const int hi   = lane >> 4;     // K-half selector for A/B operands

  const int n0 = blockIdx.x * 16;       // packed-batch row base (chunk)
  const int j  = n0 / B_SZ;             // dilation phase of this chunk
  const int b0 = n0 % B_SZ;             // batch base
  const int Td = T_LEN / rate;

  const int colh = wave * 16 + lo;      // owned h-column (0..255)

  float bias[4];
  const float* pgx[4];
  const float* pgh[4];
#pragma unroll
  for (int g = 0; g < 4; ++g) {
    int colg = g * 256 + colh;          // column in 4H gate space (i,f,g,o)
    bias[g] = bih[colg] + bhh[colg];
    pgx[g]  = wpx + (size_t)colg * 2;
    pgh[g]  = wph + (size_t)colg * 2;
  }

  // This thread's slice of the x tile (8 consecutive floats of one row).
  const int xm = (tid * 8) >> 8;        // tile row 0..15
  const int xc = (tid * 8) & 255;       // tile col, multiple of 8
  // Raw LDS byte offset of our slice (addrspace(3) ptrtoint -> LDS offset).
  const unsigned lds_dst =
      (unsigned)(uintptr_t)(__attribute__((address_space(3))) float*)&x_lds[xm][xc];

  // h0 = 0
  for (int i = tid; i < 16 * 256; i += 512) h_lds[i >> 8][i & 255] = 0.0f;
  v8f c8;                                // c0 = 0 (16x16 tile, 8 elems/lane)
#pragma unroll
  for (int e = 0; e < 8; ++e) c8[e] = 0.0f;
  __syncthreads();

  for (int t = 0; t < Td; ++t) {
    const int tg = t * rate + j;        // global timestep this chunk maps to

    // ---- DMA x tile into LDS: 16 contiguous (b) rows x 256 features ----
    {
      const float* gsrc = xin + ((size_t)tg * B_SZ + b0) * 256 + xm * 256 + xc;
#pragma unroll
      for (int q = 0; q < 4; ++q) {
        asm volatile("global_load_async_to_lds_b64 %0, %1, off"
                     :: "v"(lds_dst + q * 8u), "v"(gsrc + 2 * q)
                     : "memory");
      }
      if (t + 1 < Td) {   // uniform branch; warm L2 for next step's tile
        __builtin_prefetch(xin + ((size_t)(tg + rate) * B_SZ + b0) * 256 + tid * 8, 0, 0);
      }
      asm volatile("s_wait_asynccnt 0" ::: "memory");  // our LDS writes landed
    }
    __syncthreads();

    // ---- gates = bias + x@Wih^T + h@Whh^T via FP32 WMMA ----
    v8f acc[4];
#pragma unroll
    for (int g = 0; g < 4; ++g)
#pragma unroll
      for (int e = 0; e < 8; ++e) acc[g][e] = bias[g];

#pragma unroll 2
    for (int k0 = 0; k0 < 256; k0 += 4) {
      const int kk   = k0 + 2 * hi;        // this lane's K pair
      const int pidx = (k0 >> 1) + hi;     // packed-weight pair index
      v2f ax = *(const v2f*)&x_lds[lo][kk];
      v2f ah = *(const v2f*)&h_lds[lo][kk];
#pragma unroll
      for (int g = 0; g < 4; ++g) {
        v2f bx = *(const v2f*)(pgx[g] + (size_t)pidx * 2048);
        v2f bh = *(const v2f*)(pgh[g] + (size_t)pidx * 2048);
        acc[g] = __builtin_amdgcn_wmma_f32_16x16x4_f32(
            false, ax, false, bx, (short)0, acc[g], false, false);
        acc[g] = __builtin_amdgcn_wmma_f32_16x16x4_f32(
            false, ah, false, bh, (short)0, acc[g], false, false);
      }
    }

    // ---- wave-local LSTM cell update (owns all 4 gates for its columns) ----
    float hreg[8];
#pragma unroll
    for (int e = 0; e < 8; ++e) {
      float iv = fast_sigmoid(acc[0][e]);
      float fv = fast_sigmoid(acc[1][e]);
      float gv = fast_tanh(acc[2][e]);
      float ov = fast_sigmoid(acc[3][e]);
      float cv = fv * c8[e] + iv * gv;
      c8[e] = cv;
      hreg[e] = ov * fast_tanh(cv);
    }

    __syncthreads();   // everyone finished reading old h before overwrite
    {
      float* orow = out + ((size_t)tg * B_SZ + b0) * H_SZ + colh;
#pragma unroll
      for (int e = 0; e < 8; ++e) {
        int m = e + 8 * hi;              // C/D layout: VGPR e -> row e (+8 hi)
        h_lds[m][colh] = hreg[e];
        orow[(size_t)m * H_SZ] = hreg[e];
      }
    }
    __syncthreads();   // new h visible to all waves before next step
  }
}

extern "C" void kernel_launch(void* const* d_in, const int* in_sizes, int n_in,
                              void* d_out, int out_size, void* d_ws, size_t ws_size,
                              hipStream_t stream) {
  const float* x   = (const float*)d_in[0];
  const float* Wih = (const float*)d_in[1];
  const float* Whh = (const float*)d_in[2];
  const float* bih = (const float*)d_in[3];
  const float* bhh = (const float*)d_in[4];
  float* out = (float*)d_out;
  float* wp  = (float*)d_ws;     // 6 * 262144 floats = 6 MB packed weights

  // Pack all 6 weight matrices into WMMA-B layout (L2-resident afterwards).
  wt_pack_kernel<<<dim3(6 * MAT_ELEMS / 256), dim3(256), 0, stream>>>(Wih, Whh, wp);

  const int rates[NLAYER] = {1, 2, 4};
  for (int l = 0; l < NLAYER; ++l) {
    const float* in = (l == 0) ? x : (out + (size_t)(l - 1) * T_LEN * B_SZ * H_SZ);
    const int nchunks = rates[l] * B_SZ / 16;   // 8, 16, 32 blocks
    dlstm_scan_kernel<<<dim3(nchunks), dim3(512), 0, stream>>>(
        in,
        wp + (size_t)(2 * l + 0) * MAT_ELEMS,
        wp + (size_t)(2 * l + 1) * MAT_ELEMS,
        bih + (size_t)l * G4H,
        bhh + (size_t)l * G4H,
        out + (size_t)l * T_LEN * B_SZ * H_SZ,
        rates[l]);
  }
}